// Energy_Distance_30021821399466
// MI455X (gfx1250) — compile-verified
//
#include <hip/hip_runtime.h>
#include <stdint.h>

// Energy distance, MI455X (gfx1250, wave32).
// B=16, N=M=256, D=128, f32. VALU-bound (~0.8 G ops, 4 MB data, L2-resident).
// Exact f32: |diff| computed on VALU, accumulation done by V_WMMA_F32_16X16X4_F32
// with an all-ones B matrix (column redundancy removed by exact /16).
// Tiles staged to LDS with global_load_async_to_lds_b128 (ASYNCcnt).

typedef __attribute__((ext_vector_type(2))) float v2f;
typedef __attribute__((ext_vector_type(8))) float v8f;

#define D_DIM  128
#define NR     256
#define TILE   16
#define NTILES 16   // 256 / 16
#define WAVES  8    // 256 threads / wave32

__global__ __launch_bounds__(256) void ed_partial(const float* __restrict__ x1,
                                                  const float* __restrict__ x2,
                                                  float* __restrict__ partial) {
  __shared__ float s_left[TILE * D_DIM];           // 8 KB, shared by block
  __shared__ float s_right[WAVES][TILE * D_DIM];   // 8 KB per wave = 64 KB
  __shared__ float s_wsum[WAVES];

  const int tid  = threadIdx.x;
  const int wave = tid >> 5;
  const int lane = tid & 31;
  const int hi   = lane >> 4;   // A-fragment K half: lanes 0-15 -> K=0,1; 16-31 -> K=2,3
  const int nrow = lane & 15;   // A-fragment row M

  const int ltile = blockIdx.x; // 0..15 : 16-row tile of the "left" matrix
  const int type  = blockIdx.y; // 0 = cross(x1,x2), 1 = same(x2,x2), 2 = same(x1,x1)
  const int b     = blockIdx.z; // batch

  const float* base1 = x1 + (size_t)b * NR * D_DIM;
  const float* base2 = x2 + (size_t)b * NR * D_DIM;
  const float* left  = (type == 1) ? base2 : base1;
  const float* right = (type == 2) ? base1 : base2;
  const float  w     = (type == 0) ? 1.0f : -0.5f;

  // ---- stage left tile: block-cooperative async global->LDS (512 x b128) ----
  {
    const float*   g  = left + ltile * TILE * D_DIM;
    const uint32_t l0 = (uint32_t)(uintptr_t)&s_left[0];
    for (int i = tid; i < (TILE * D_DIM) / 4; i += 256) {
      uint32_t la = l0 + i * 16;
      uint64_t ga = (uint64_t)(uintptr_t)(g + i * 4);
      asm volatile("global_load_async_to_lds_b128 %0, %1, off"
                   :: "v"(la), "v"(ga) : "memory");
    }
    asm volatile("s_wait_asynccnt 0" ::: "memory");
  }
  __syncthreads();

  v8f acc[4] = {};                 // 4 rotating WMMA accumulators (break C-chain)
  v2f bones; bones[0] = 1.0f; bones[1] = 1.0f;   // all-ones B (layout-agnostic)
  const uint32_t r0 = (uint32_t)(uintptr_t)&s_right[wave][0];

  for (int rt = wave; rt < NTILES; rt += WAVES) {   // 2 right tiles per wave
    // ---- stage this wave's right tile (wave-private LDS slot) ----
    const float* g = right + rt * TILE * D_DIM;
    for (int i = lane; i < (TILE * D_DIM) / 4; i += 32) {
      uint32_t la = r0 + i * 16;
      uint64_t ga = (uint64_t)(uintptr_t)(g + i * 4);
      asm volatile("global_load_async_to_lds_b128 %0, %1, off"
                   :: "v"(la), "v"(ga) : "memory");
    }
    asm volatile("s_wait_asynccnt 0" ::: "memory");

    // ---- accumulate 16x16 pair tile: K-chunks of 4 d-values via WMMA ----
    for (int c = 0; c < D_DIM / 4; ++c) {
      const int d0 = 4 * c + 2 * hi;               // this lane's two d's (contiguous)
      v2f l = *(const v2f*)&s_left[nrow * D_DIM + d0];
#pragma unroll
      for (int mp = 0; mp < 16; ++mp) {
        v2f r = *(const v2f*)&s_right[wave][mp * D_DIM + d0];
        v2f a;
        a[0] = __builtin_fabsf(l[0] - r[0]);
        a[1] = __builtin_fabsf(l[1] - r[1]);
        // D[n,j] += sum_k A[n,k] (B=ones); exact f32 accumulation on the matrix pipe
        acc[mp & 3] = __builtin_amdgcn_wmma_f32_16x16x4_f32(
            false, a, false, bones, (short)0, acc[mp & 3], false, false);
      }
    }
  }

  // ---- deterministic reduction: lanes -> wave -> block -> partial[] ----
  v8f t = (acc[0] + acc[1]) + (acc[2] + acc[3]);
  float s = ((t[0] + t[1]) + (t[2] + t[3])) + ((t[4] + t[5]) + (t[6] + t[7]));
#pragma unroll
  for (int off = 16; off >= 1; off >>= 1)
    s += __shfl_xor(s, off, 32);
  if (lane == 0) s_wsum[wave] = s;
  __syncthreads();
  if (tid == 0) {
    float bs = 0.0f;
#pragma unroll
    for (int i = 0; i < WAVES; ++i) bs += s_wsum[i];
    // /65536 = pair mean; /16 = ones-B column redundancy (both exact powers of 2)
    partial[((size_t)b * 3 + type) * NTILES + ltile] =
        w * bs * (1.0f / (65536.0f * 16.0f));
  }
}

__global__ void ed_reduce(const float* __restrict__ partial,
                          float* __restrict__ out) {
  const int b = threadIdx.x;   // 16 threads, one per batch
  float s = 0.0f;
  for (int i = 0; i < 3 * NTILES; ++i)
    s += partial[b * 3 * NTILES + i];
  out[b] = s;
}

extern "C" void kernel_launch(void* const* d_in, const int* in_sizes, int n_in,
                              void* d_out, int out_size, void* d_ws, size_t ws_size,
                              hipStream_t stream) {
  const float* x1 = (const float*)d_in[0];   // [16, 256, 128] f32
  const float* x2 = (const float*)d_in[1];   // [16, 256, 128] f32
  float* out      = (float*)d_out;           // [16] f32
  float* partial  = (float*)d_ws;            // 16*3*16 = 768 floats scratch

  dim3 grid(NTILES, 3, 16);                  // tiles x pair-types x batches
  ed_partial<<<grid, 256, 0, stream>>>(x1, x2, partial);
  ed_reduce<<<1, 16, 0, stream>>>(partial, out);
}